// InterferometricWaveGating_89644557402523
// MI455X (gfx1250) — compile-verified
//
#include <hip/hip_runtime.h>

// Problem constants (from reference)
#define BB 4
#define LL 128
#define DD 256
#define NN 8192          // L * NUM_KNOTS
#define HD 4             // head dim
// SCALE * log2(e) folded into Q: 0.5 * 1.4426950408889634
#define QSCALE 0.72134752044448170f

typedef __attribute__((ext_vector_type(2))) float v2f;
typedef __attribute__((ext_vector_type(4))) float v4f;
typedef __attribute__((ext_vector_type(8))) float v8f;

// ---- CDNA5 feature detection (device pass only) ---------------------------
#if defined(__gfx1250__)
#if __has_builtin(__builtin_amdgcn_tensor_load_to_lds)
#define HAVE_TDM 1
#endif
#if __has_builtin(__builtin_amdgcn_global_load_async_to_lds_b128)
#define HAVE_ASYNC 1
#endif
#endif

#if __has_include(<hip/amd_detail/amd_gfx1250_TDM.h>)
#define TDM_6ARG 1   // therock toolchain: 6-arg tensor builtin
#endif

// D = A(16x4) * B(4x16) + C, all f32, wave32.
// A layout: lane l holds row M=l&15; VGPR v holds K = v + 2*(l>>4).
// B layout (mirror): lane l holds col N=l&15; VGPR v holds K = v + 2*(l>>4).
// C/D layout: lane l, VGPR r -> M = r + 8*(l>>4), N = l&15.
__device__ __forceinline__ v8f wmma16x16x4(v2f a, v2f b, v8f c) {
  return __builtin_amdgcn_wmma_f32_16x16x4_f32(false, a, false, b, (short)0, c,
                                               false, false);
}

// lane <-> lane^16 exchange via ds_swizzle (group-of-32, xor=0x10, and=0x1f)
__device__ __forceinline__ float swap16(float x) {
  return __int_as_float(__builtin_amdgcn_ds_swizzle(__float_as_int(x), 0x401F));
}

#if defined(HAVE_TDM)
typedef unsigned int tdm_v4u __attribute__((ext_vector_type(4)));
typedef int tdm_v8i __attribute__((ext_vector_type(8)));
typedef int tdm_v4i __attribute__((ext_vector_type(4)));

// 1-D contiguous global->LDS tensor DMA: nelem f32 elements.
// D# per CDNA5 ISA ch.8: group0 = {count/flags, lds_addr, global_addr, type=2},
// group1 = {data_size=4B, tensor_dim0=tile_dim0=nelem, dims1=1, stride0=nelem}.
__device__ __forceinline__ void tdm_load_1d(const float* gsrc,
                                            unsigned lds_byte_off,
                                            unsigned nelem) {
  unsigned long long ga = (unsigned long long)(const void*)gsrc;
  tdm_v4u g0;
  g0.x = 1u;                                    // count=1, user descriptor
  g0.y = lds_byte_off;                          // lds_addr [63:32]
  g0.z = (unsigned)(ga & 0xFFFFFFFFu);          // global_addr [95:64]
  g0.w = (unsigned)((ga >> 32) & 0x01FFFFFFu)   // global_addr [120:96]
         | (2u << 30);                          // type=2 ("image") [127:126]
  tdm_v8i g1;
  g1[0] = (int)(2u << 16);                      // data_size=4B, wg_mask=0
  g1[1] = (int)((nelem & 0xFFFFu) << 16);       // tensor_dim0[15:0] @ bit48
  g1[2] = (int)((nelem >> 16) | (1u << 16));    // tensor_dim0[31:16] | tdim1=1
  g1[3] = (int)((nelem & 0xFFFFu) << 16);       // tile_dim0 @ bits[127:112]
  g1[4] = 1;                                    // tile_dim1=1, tile_dim2=0
  g1[5] = (int)nelem;                           // tensor_dim0_stride[31:0]
  g1[6] = 0;
  g1[7] = 0;
  tdm_v4i z4 = {0, 0, 0, 0};
#if defined(TDM_6ARG)
  tdm_v8i z8 = {0, 0, 0, 0, 0, 0, 0, 0};
  __builtin_amdgcn_tensor_load_to_lds(g0, g1, z4, z4, z8, 0);
#else
  __builtin_amdgcn_tensor_load_to_lds(g0, g1, z4, z4, 0);
#endif
}

__device__ __forceinline__ void wait_tensorcnt0() {
#if __has_builtin(__builtin_amdgcn_s_wait_tensorcnt)
  __builtin_amdgcn_s_wait_tensorcnt(0);
#else
  asm volatile("s_wait_tensorcnt 0x0" ::: "memory");
#endif
}
#endif  // HAVE_TDM

#if defined(HAVE_ASYNC)
// Builtin signature (from clang diagnostic): (as1 int4*, as3 int4*, Imm, Imm)
// with gcc-style vector_size(16) int4 pointees.
typedef int gv4i __attribute__((vector_size(16)));
__device__ __forceinline__ void async_cp16(const float* g, float* lds) {
  __builtin_amdgcn_global_load_async_to_lds_b128(
      (__attribute__((address_space(1))) gv4i*)g,
      (__attribute__((address_space(3))) gv4i*)lds, 0, 0);
}
__device__ __forceinline__ void wait_asynccnt0() {
#if __has_builtin(__builtin_amdgcn_s_wait_asynccnt)
  __builtin_amdgcn_s_wait_asynccnt(0);
#else
  asm volatile("s_wait_asynccnt 0x0" ::: "memory");
#endif
}
#endif  // HAVE_ASYNC

// ---------------------------------------------------------------------------
// Kernel 1: Q/K/V projections.  C[m,n] = sum_k x[m,k] * W[n,k]
// One wave per 16x16 output tile; 3 projections * 512 tiles = 1536 waves.
// ---------------------------------------------------------------------------
__global__ __launch_bounds__(256) void qkv_proj_kernel(
    const float* __restrict__ x, const float* __restrict__ Wq,
    const float* __restrict__ Wk, const float* __restrict__ Wv,
    float* __restrict__ qbuf, float* __restrict__ kbuf,
    float* __restrict__ vbuf) {
  const int lane = threadIdx.x & 31;
  const int wave = threadIdx.x >> 5;
  const int w = blockIdx.x * 8 + wave;  // 0..1535
  const int proj = w >> 9;              // /512
  const int t = w & 511;
  const int m0 = (t >> 4) << 4;
  const int n0 = (t & 15) << 4;
  const float* Wm = (proj == 0) ? Wq : (proj == 1) ? Wk : Wv;
  float* outp = (proj == 0) ? qbuf : (proj == 1) ? kbuf : vbuf;

  const int l15 = lane & 15;
  const int kh = (lane >> 4) << 1;  // 0 or 2
  const float* arow = x + (m0 + l15) * DD;   // A row m, k contiguous
  const float* brow = Wm + (n0 + l15) * DD;  // B col n == W row n, k contiguous

  v8f acc = {0.f, 0.f, 0.f, 0.f, 0.f, 0.f, 0.f, 0.f};
#pragma unroll 8
  for (int k0 = 0; k0 < DD; k0 += 4) {
    v2f a = *(const v2f*)(arow + k0 + kh);
    v2f b = *(const v2f*)(brow + k0 + kh);
    acc = wmma16x16x4(a, b, acc);
  }
  const int mrow = m0 + (lane >> 4) * 8;
#pragma unroll
  for (int r = 0; r < 8; ++r)
    outp[(mrow + r) * DD + n0 + l15] = acc[r];
}

// ---------------------------------------------------------------------------
// Kernel 2: flash attention over N=8192 knots, head_dim=4.
// Whole K,V for one batch staged in LDS (2 * 128 KB of the 320 KB WGP LDS):
// K via Tensor Data Mover DMA, V via async global->LDS b128 (with fallbacks).
// Each wave owns one 16-row Q tile; computes S^T tiles via WMMA f32 16x16x4,
// online softmax in exp2 domain, PV accumulated per-lane from LDS.
// ---------------------------------------------------------------------------
__global__ __launch_bounds__(256) void attn_kernel(
    const float* __restrict__ qbuf, const float* __restrict__ kbuf,
    const float* __restrict__ vbuf, float* __restrict__ abuf) {
  __shared__ __align__(16) float ldsK[NN * HD];  // 128 KB
  __shared__ __align__(16) float ldsV[NN * HD];  // 128 KB

  const int b = blockIdx.x >> 6;     // batch
  const int qblk = blockIdx.x & 63;  // 8 q-tiles per block
  const int tid = threadIdx.x;
  const int lane = tid & 31;
  const int wave = tid >> 5;

  const float* kg = kbuf + b * (NN * HD);
  const float* vg = vbuf + b * (NN * HD);

  // ---- stage K,V for this batch into LDS --------------------------------
#if defined(HAVE_TDM) && defined(HAVE_ASYNC)
  if (wave == 0) {  // one TDM descriptor moves all of K (EXEC ignored)
    tdm_load_1d(kg, (unsigned)(unsigned long long)(uintptr_t)ldsK, NN * HD);
  }
  for (int i = tid * 4; i < NN * HD; i += 256 * 4)  // V: async b128 per lane
    async_cp16(vg + i, ldsV + i);
  wait_asynccnt0();
  if (wave == 0) wait_tensorcnt0();
#elif defined(HAVE_TDM)
  if (wave == 0) {
    tdm_load_1d(kg, (unsigned)(unsigned long long)(uintptr_t)ldsK, NN * HD);
    tdm_load_1d(vg, (unsigned)(unsigned long long)(uintptr_t)ldsV, NN * HD);
    wait_tensorcnt0();
  }
#elif defined(HAVE_ASYNC)
  for (int i = tid * 4; i < NN * HD; i += 256 * 4) {
    async_cp16(kg + i, ldsK + i);
    async_cp16(vg + i, ldsV + i);
  }
  wait_asynccnt0();
#else
  for (int i = tid * 4; i < NN * HD; i += 256 * 4) {
    *(v4f*)(ldsK + i) = *(const v4f*)(kg + i);
    *(v4f*)(ldsV + i) = *(const v4f*)(vg + i);
  }
#endif
  __syncthreads();

  const int q0 = (qblk * 8 + wave) * 16;
  const int l15 = lane & 15;
  const int hi = lane >> 4;  // 0/1: which K-pair / which M-half of C

  // Q tile in B-layout, pre-scaled by SCALE*log2e.
  v2f qb = *(const v2f*)(qbuf + (b * NN + q0 + l15) * HD + hi * 2);
  qb *= QSCALE;

  float m = -1e30f, lsum = 0.f;
  float o0 = 0.f, o1 = 0.f, o2 = 0.f, o3 = 0.f;
  const v8f zero = {0.f, 0.f, 0.f, 0.f, 0.f, 0.f, 0.f, 0.f};

#pragma unroll 2
  for (int kt = 0; kt < NN; kt += 16) {
    // A = K tile rows (M = k), K-dim = head dim.
    v2f ka = *(const v2f*)(ldsK + (kt + l15) * HD + hi * 2);
    v8f s = wmma16x16x4(ka, qb, zero);  // S^T[k, q] (already * scale * log2e)

    // tile max over 16 k rows of this lane's q column
    float tm = fmaxf(fmaxf(fmaxf(s[0], s[1]), fmaxf(s[2], s[3])),
                     fmaxf(fmaxf(s[4], s[5]), fmaxf(s[6], s[7])));
    tm = fmaxf(tm, swap16(tm));
    const float mn = fmaxf(m, tm);
    const float cf = __builtin_amdgcn_exp2f(m - mn);
    o0 *= cf; o1 *= cf; o2 *= cf; o3 *= cf; lsum *= cf;
    m = mn;

    const int kbase = kt + hi * 8;  // this lane's 8 k rows
#pragma unroll
    for (int r = 0; r < 8; ++r) {
      const float p = __builtin_amdgcn_exp2f(s[r] - mn);
      lsum += p;
      const float* vv = ldsV + (kbase + r) * HD;  // broadcast across 16 lanes
      o0 = fmaf(p, vv[0], o0);
      o1 = fmaf(p, vv[1], o1);
      o2 = fmaf(p, vv[2], o2);
      o3 = fmaf(p, vv[3], o3);
    }
  }

  // Combine the two lane-halves (same q column, disjoint k halves; m already
  // synchronized across halves every tile).
  o0 += swap16(o0); o1 += swap16(o1); o2 += swap16(o2); o3 += swap16(o3);
  lsum += swap16(lsum);
  const float inv = 1.f / lsum;
  if (hi == 0) {
    v4f o = {o0 * inv, o1 * inv, o2 * inv, o3 * inv};
    *(v4f*)(abuf + (b * NN + q0 + l15) * HD) = o;
  }
}

// ---------------------------------------------------------------------------
// Kernel 3: output projection  out = attn_out @ Wo^T + bo
// ---------------------------------------------------------------------------
__global__ __launch_bounds__(256) void out_proj_kernel(
    const float* __restrict__ abuf, const float* __restrict__ Wo,
    const float* __restrict__ bo, float* __restrict__ out) {
  const int lane = threadIdx.x & 31;
  const int wave = threadIdx.x >> 5;
  const int t = blockIdx.x * 8 + wave;  // 0..511
  const int m0 = (t >> 4) << 4;
  const int n0 = (t & 15) << 4;
  const int l15 = lane & 15;
  const int kh = (lane >> 4) << 1;
  const float* arow = abuf + (m0 + l15) * DD;
  const float* brow = Wo + (n0 + l15) * DD;

  v8f acc = {0.f, 0.f, 0.f, 0.f, 0.f, 0.f, 0.f, 0.f};
#pragma unroll 8
  for (int k0 = 0; k0 < DD; k0 += 4) {
    v2f a = *(const v2f*)(arow + k0 + kh);
    v2f b = *(const v2f*)(brow + k0 + kh);
    acc = wmma16x16x4(a, b, acc);
  }
  const float bias = bo[n0 + l15];
  const int mrow = m0 + (lane >> 4) * 8;
#pragma unroll
  for (int r = 0; r < 8; ++r)
    out[(mrow + r) * DD + n0 + l15] = acc[r] + bias;
}

extern "C" void kernel_launch(void* const* d_in, const int* in_sizes, int n_in,
                              void* d_out, int out_size, void* d_ws,
                              size_t ws_size, hipStream_t stream) {
  const float* x = (const float*)d_in[0];
  const float* Wq = (const float*)d_in[1];
  const float* Wk = (const float*)d_in[2];
  const float* Wv = (const float*)d_in[3];
  const float* Wo = (const float*)d_in[4];
  const float* bo = (const float*)d_in[5];

  float* ws = (float*)d_ws;
  const int seg = BB * NN * HD;  // 131072 floats = 512 KB
  float* qbuf = ws;
  float* kbuf = ws + seg;
  float* vbuf = ws + 2 * seg;
  float* abuf = ws + 3 * seg;  // total 2 MB of workspace

  qkv_proj_kernel<<<192, 256, 0, stream>>>(x, Wq, Wk, Wv, qbuf, kbuf, vbuf);
  attn_kernel<<<256, 256, 0, stream>>>(qbuf, kbuf, vbuf, abuf);
  out_proj_kernel<<<64, 256, 0, stream>>>(abuf, Wo, bo, (float*)d_out);
}